// LightGCN_74156905333132
// MI455X (gfx1250) — compile-verified
//
#include <hip/hip_runtime.h>

#define EMBED_DIM 64
#define TILE 256
#define BLOCK 256
#define MAX_SPMM_BLOCKS 4096

#if __has_builtin(__builtin_amdgcn_global_load_async_to_lds_b32)
#define HAVE_ASYNC_LDS 1
#else
#define HAVE_ASYNC_LDS 0
#endif

// Builtin prototype (confirmed by hipcc diagnostic):
//   __builtin_amdgcn_global_load_async_to_lds_b32(global int* src, shared int* dst, imm offset, imm cpol)
typedef __attribute__((address_space(1))) int g_b32_t;
typedef __attribute__((address_space(3))) int l_b32_t;

__device__ __forceinline__ void async_wait_le(int pending) {
#if HAVE_ASYNC_LDS
    if (pending) {
#if __has_builtin(__builtin_amdgcn_s_wait_asynccnt)
        __builtin_amdgcn_s_wait_asynccnt(3);
#else
        asm volatile("s_wait_asynccnt 3" ::: "memory");
#endif
    } else {
#if __has_builtin(__builtin_amdgcn_s_wait_asynccnt)
        __builtin_amdgcn_s_wait_asynccnt(0);
#else
        asm volatile("s_wait_asynccnt 0" ::: "memory");
#endif
    }
#else
    (void)pending;
#endif
}

// Four no-return fp32 global atomics to one 16B-aligned row segment.
// Inline asm guarantees native GLOBAL_ATOMIC_ADD_F32 (STOREcnt only, adds at L2)
// instead of a CAS expansion.
__device__ __forceinline__ void atomic_add4_f32(float* p, float a, float b, float c, float d) {
    unsigned long long addr = (unsigned long long)(size_t)p;
    asm volatile(
        "global_atomic_add_f32 %0, %1, off\n\t"
        "global_atomic_add_f32 %0, %2, off offset:4\n\t"
        "global_atomic_add_f32 %0, %3, off offset:8\n\t"
        "global_atomic_add_f32 %0, %4, off offset:12"
        :: "v"(addr), "v"(a), "v"(b), "v"(c), "v"(d)
        : "memory");
}

// COO SpMM: y[row] += val * x[col]. Persistent blocks, grid-stride over
// 256-nnz tiles. COO triples are streamed into LDS with CDNA5 async
// global->LDS copies, double-buffered so the next tile's stream overlaps the
// current tile's gather + atomic scatter. 16 lanes cover one 64-float row
// (float4 each). x and y both fit in the 192MB L2 together.
extern "C" __global__ __launch_bounds__(BLOCK)
void lg_spmm(const int* __restrict__ rows, const int* __restrict__ cols,
             const float* __restrict__ vals, const float* __restrict__ x,
             float* __restrict__ y, int nnz) {
    __shared__ int   s_row[2][TILE];
    __shared__ int   s_col[2][TILE];
    __shared__ float s_val[2][TILE];

    const int tid      = threadIdx.x;
    const int stride   = (int)gridDim.x;
    const int numTiles = (nnz + TILE - 1) / TILE;

    // Stage tile t's COO triples into LDS buffer b (this thread's element).
    auto stage = [&](int t, int b) {
        const int base = t * TILE;
        const int cnt  = min(nnz - base, TILE);
        if (tid < cnt) {
#if HAVE_ASYNC_LDS
            __builtin_amdgcn_global_load_async_to_lds_b32(
                (g_b32_t*)(rows + base + tid), (l_b32_t*)&s_row[b][tid], 0, 0);
            __builtin_amdgcn_global_load_async_to_lds_b32(
                (g_b32_t*)(cols + base + tid), (l_b32_t*)&s_col[b][tid], 0, 0);
            __builtin_amdgcn_global_load_async_to_lds_b32(
                (g_b32_t*)(vals + base + tid), (l_b32_t*)&s_val[b][tid], 0, 0);
#else
            s_row[b][tid] = rows[base + tid];
            s_col[b][tid] = cols[base + tid];
            s_val[b][tid] = vals[base + tid];
#endif
        }
    };

    int t = (int)blockIdx.x;
    if (t >= numTiles) return;

    int buf = 0;
    stage(t, buf);                               // prologue: stream first tile

    const int sub = tid & 15;                    // which float4 of the 64-float row
    for (; t < numTiles; t += stride) {
        const int  tn       = t + stride;
        const bool haveNext = tn < numTiles;
        if (haveNext) stage(tn, buf ^ 1);        // overlap next tile's stream

        async_wait_le(haveNext);                 // older 3 async ops (this tile) done
        __syncthreads();                         // whole tile visible to all waves

        const int cnt = min(nnz - t * TILE, TILE);
        for (int i = tid >> 4; i < cnt; i += BLOCK / 16) {
            const int   c = s_col[buf][i];       // LDS broadcast across 16 lanes
            const int   r = s_row[buf][i];
            const float v = s_val[buf][i];
            const float4 g = *(const float4*)(x + (size_t)c * EMBED_DIM + sub * 4);
            atomic_add4_f32(y + (size_t)r * EMBED_DIM + sub * 4,
                            g.x * v, g.y * v, g.z * v, g.w * v);
        }
        __syncthreads();                         // reads done before buffer reuse
        buf ^= 1;
    }
}

// acc = concat(user, item); cur = same. 128-bit vectorized.
extern "C" __global__ __launch_bounds__(BLOCK)
void lg_init(const float* __restrict__ ue, const float* __restrict__ ie,
             float* __restrict__ acc, float* __restrict__ cur,
             long long userElems, long long totalElems) {
    long long i4 = (blockIdx.x * (long long)BLOCK + threadIdx.x) * 4;
    if (i4 >= totalElems) return;
    const float4 v = (i4 < userElems)
        ? *(const float4*)(ue + i4)
        : *(const float4*)(ie + (i4 - userElems));
    *(float4*)(acc + i4) = v;
    *(float4*)(cur + i4) = v;
}

// acc += cur, 128-bit vectorized.
extern "C" __global__ __launch_bounds__(BLOCK)
void lg_axpy(float* __restrict__ acc, const float* __restrict__ cur, long long n4) {
    long long i = blockIdx.x * (long long)BLOCK + threadIdx.x;
    if (i >= n4) return;
    float4*       a = (float4*)acc + i;
    const float4  b = *((const float4*)cur + i);
    float4 t = *a;
    t.x += b.x; t.y += b.y; t.z += b.z; t.w += b.w;
    *a = t;
}

extern "C" void kernel_launch(void* const* d_in, const int* in_sizes, int n_in,
                              void* d_out, int out_size, void* d_ws, size_t ws_size,
                              hipStream_t stream) {
    (void)n_in; (void)ws_size;
    const float* adj_vals = (const float*)d_in[0];
    const float* ue       = (const float*)d_in[1];
    const float* ie       = (const float*)d_in[2];
    const int*   rows     = (const int*)d_in[3];
    const int*   cols     = (const int*)d_in[4];

    const int       nnz        = in_sizes[0];
    const long long userElems  = in_sizes[1];
    const long long totalElems = out_size;          // N_NODES * 64
    const long long n4         = totalElems / 4;

    float* acc  = (float*)d_out;
    float* buf0 = (float*)d_ws;                     // cur / next ping-pong
    float* buf1 = buf0 + totalElems;                // needs 2*totalElems*4 bytes

    const unsigned initBlocks = (unsigned)((n4 + BLOCK - 1) / BLOCK);
    const int      numTiles   = (nnz + TILE - 1) / TILE;
    const unsigned spmmBlocks = (unsigned)(numTiles < MAX_SPMM_BLOCKS ? numTiles : MAX_SPMM_BLOCKS);

    lg_init<<<initBlocks, BLOCK, 0, stream>>>(ue, ie, acc, buf0, userElems, totalElems);

    float* cur = buf0;
    float* nxt = buf1;
    for (int layer = 0; layer < 3; ++layer) {
        (void)hipMemsetAsync(nxt, 0, (size_t)totalElems * sizeof(float), stream);
        lg_spmm<<<spmmBlocks, BLOCK, 0, stream>>>(rows, cols, adj_vals, cur, nxt, nnz);
        lg_axpy<<<initBlocks, BLOCK, 0, stream>>>(acc, nxt, n4);
        float* t = cur; cur = nxt; nxt = t;
    }
}